// LargeKernelBSC_S_25434796327721
// MI455X (gfx1250) — compile-verified
//
#include <hip/hip_runtime.h>

typedef float v2f __attribute__((ext_vector_type(2)));
typedef float v8f __attribute__((ext_vector_type(8)));

// ---------------------------------------------------------------------------
// Fused pointwise pair:  Z = W2 @ (W1 @ X + b1) + b2   (per batch image)
//   X: [C, HW] row-major.  W1: [MID, C].  W2: [C, MID].
// One workgroup = 8 waves = 256 threads owns NT spatial columns of one batch.
// X tile transposed into LDS as [NT][C+4] (async-to-LDS scatter), so every
// WMMA B-fragment is a single conflict-free ds_load_b64.  mid in chunks of
// 32; Y chunk transposed in LDS as [NT][36]; Z in VGPR accumulators.
// Matrix math: V_WMMA_F32_16X16X4_F32 (full fp32 precision).
// ---------------------------------------------------------------------------
template <int C, int MID, int NT>
__global__ __launch_bounds__(256, 1)
void fused_pw(const float* __restrict__ x, const float* __restrict__ w1,
              const float* __restrict__ b1, const float* __restrict__ w2,
              const float* __restrict__ b2, float* __restrict__ z, int hw)
{
    constexpr int MC  = 32;               // mid chunk
    constexpr int XPT = C + 4;            // XsT row pitch (4j bank stride)
    constexpr int YPT = MC + 4;           // YsT row pitch (36)
    constexpr int NTT = NT / 16;
    constexpr int T1  = (MC / 16) * NTT;  // step-1 tiles
    constexpr int T2  = (C  / 16) * NTT;  // step-2 tiles
    constexpr int TPW = T2 / 8;           // Z tiles per wave

    extern __shared__ float smem[];
    float* XsT = smem;                    // [NT][XPT]  X^T tile
    float* YsT = smem + NT * XPT;         // [NT][YPT]  Y^T chunk

    const int tid  = threadIdx.x;
    const int wave = tid >> 5;
    const int lane = tid & 31;
    const int lr   = lane & 15;           // row/col within 16
    const int kh   = (lane >> 4) << 1;    // K offset 0/2 (lane half)
    const int mh   = (lane >> 4) << 3;    // M offset 0/8 for C/D frags

    const int    n0    = blockIdx.x * NT;
    const size_t xbase = (size_t)blockIdx.y * C * hw + n0;

    // ---- async-DMA the X tile into LDS, transposing [C x NT] -> [NT][C] ----
    {
        const uint32_t ldsBase = (uint32_t)(uintptr_t)XsT;
        for (int i = tid; i < C * NT; i += 256) {
            const int c = i / NT, n = i % NT;
            const uint64_t ga = (uint64_t)(uintptr_t)(x + xbase + (size_t)c * hw + n);
            const uint32_t la = ldsBase + (uint32_t)(n * XPT + c) * 4u;
            asm volatile("global_load_async_to_lds_b32 %0, %1, off"
                         :: "v"(la), "v"(ga) : "memory");
        }
        asm volatile("s_wait_asynccnt 0x0" ::: "memory");
    }
    __syncthreads();

    v8f zero = {};
    v8f acc[TPW];
#pragma unroll
    for (int i = 0; i < TPW; ++i) acc[i] = zero;

    for (int mc = 0; mc < MID; mc += MC) {
        if (mc + MC < MID)  // warm next W1 chunk (global_prefetch_b8)
            __builtin_prefetch(w1 + (size_t)(mc + MC) * C, 0, 1);

        // ---- step 1: Y^T[NT x MC] = (W1[mc:mc+MC, :] @ X)^T  (+ b1) -------
        for (int t1 = wave; t1 < T1; t1 += 8) {
            const int m0  = (t1 / NTT) * 16;
            const int nn0 = (t1 % NTT) * 16;
            v8f a1 = zero;
            const float* w1row = w1 + (size_t)(mc + m0 + lr) * C + kh;
            const float* brow  = &XsT[(nn0 + lr) * XPT + kh];
#pragma unroll 2
            for (int k0 = 0; k0 < C; k0 += 16) {
                v2f a[4];
#pragma unroll
                for (int q = 0; q < 4; ++q) {       // batched A-frag loads
                    a[q].x = w1row[k0 + 4 * q];
                    a[q].y = w1row[k0 + 4 * q + 1];
                }
#pragma unroll
                for (int q = 0; q < 4; ++q) {       // 4 WMMAs per wait
                    v2f b;                          // one ds_load_b64
                    b.x = brow[k0 + 4 * q];
                    b.y = brow[k0 + 4 * q + 1];
                    a1 = __builtin_amdgcn_wmma_f32_16x16x4_f32(
                            false, a[q], false, b, (short)0, a1, false, false);
                }
            }
            float* yrow = &YsT[(nn0 + lr) * YPT + m0 + mh];
#pragma unroll
            for (int v = 0; v < 8; ++v)             // 8 consecutive -> b128
                yrow[v] = a1[v] + b1[mc + m0 + mh + v];
        }
        __syncthreads();   // Y chunk ready

        // ---- step 2: acc[C x NT] += W2[:, mc:mc+MC] @ Y -------------------
#pragma unroll
        for (int i = 0; i < TPW; ++i) {
            const int t   = wave + i * 8;
            const int c0  = (t / NTT) * 16;
            const int nn0 = (t % NTT) * 16;
            const float* w2row = w2 + (size_t)(c0 + lr) * MID + mc + kh;
            const float* brow  = &YsT[(nn0 + lr) * YPT + kh];
            v2f a[8];
#pragma unroll
            for (int q = 0; q < 8; ++q) {           // batched A-frag loads
                a[q].x = w2row[4 * q];
                a[q].y = w2row[4 * q + 1];
            }
            v8f d = acc[i];
#pragma unroll
            for (int q = 0; q < 8; ++q) {           // 8 WMMAs, one wait
                v2f b;
                b.x = brow[4 * q];
                b.y = brow[4 * q + 1];
                d = __builtin_amdgcn_wmma_f32_16x16x4_f32(
                        false, a[q], false, b, (short)0, d, false, false);
            }
            acc[i] = d;
        }
        __syncthreads();   // before step 1 rewrites YsT
    }

    // ---- store Z = acc + b2 ------------------------------------------------
#pragma unroll
    for (int i = 0; i < TPW; ++i) {
        const int t   = wave + i * 8;
        const int c0  = (t / NTT) * 16;
        const int nn0 = (t % NTT) * 16;
#pragma unroll
        for (int v = 0; v < 8; ++v) {
            const int c = c0 + v + mh;
            z[xbase + (size_t)c * hw + nn0 + lr] = acc[i][v] + b2[c];
        }
    }
}

// ---------------------------------------------------------------------------
// 21x21 depthwise conv (pad 10) + channel bias + residual.
// One block = one 16x16 output tile of one (b, c) plane. Haloed 36x36 input
// tile + 441 weights staged in LDS; 441 fp32 FMAs per output element.
// ---------------------------------------------------------------------------
__global__ __launch_bounds__(256, 1)
void dwconv21(const float* __restrict__ z, const float* __restrict__ xin,
              const float* __restrict__ wk, const float* __restrict__ db,
              float* __restrict__ out, int C, int H)
{
    __shared__ float tile[36 * 36];
    __shared__ float w[441];

    const int tid    = threadIdx.x;
    const int tilesX = (H + 15) >> 4;
    const int tx     = blockIdx.x % tilesX;
    const int ty     = blockIdx.x / tilesX;
    const int c      = blockIdx.y;
    const int b      = blockIdx.z;
    const size_t base = ((size_t)b * C + c) * (size_t)(H * H);
    const int oy0 = ty * 16, ox0 = tx * 16;

    for (int i = tid; i < 441; i += 256) w[i] = wk[c * 441 + i];
    for (int i = tid; i < 36 * 36; i += 256) {
        const int r  = i / 36, q = i % 36;
        const int gy = oy0 - 10 + r, gx = ox0 - 10 + q;
        float v = 0.0f;
        if (gy >= 0 && gy < H && gx >= 0 && gx < H)
            v = z[base + gy * H + gx];
        tile[i] = v;
    }
    __syncthreads();

    const int r0 = tid >> 4, c0 = tid & 15;
    const int oy = oy0 + r0, ox = ox0 + c0;
    if (oy < H && ox < H) {
        float acc = db[c];
#pragma unroll 3
        for (int ky = 0; ky < 21; ++ky) {
            const float* trow = &tile[(r0 + ky) * 36 + c0];
            const float* wrow = &w[ky * 21];
#pragma unroll
            for (int kx = 0; kx < 21; ++kx)
                acc = fmaf(trow[kx], wrow[kx], acc);
        }
        const size_t idx = base + oy * H + ox;
        out[idx] = acc + xin[idx];
    }
}

// ---------------------------------------------------------------------------
// Launch: per level l, inputs are d_in[l]=x, params flat at d_in[5+6l..].
// Z intermediate reused in workspace (max 33.5 MB, stays hot in 192MB L2).
// ---------------------------------------------------------------------------
extern "C" void kernel_launch(void* const* d_in, const int* in_sizes, int n_in,
                              void* d_out, int out_size, void* d_ws, size_t ws_size,
                              hipStream_t stream)
{
    (void)in_sizes; (void)n_in; (void)out_size; (void)ws_size;
    float* zws = (float*)d_ws;
    size_t outOff = 0;

#define RUN_LEVEL(L, CC, HH, MM, NN)                                           \
    do {                                                                       \
        const float* x  = (const float*)d_in[L];                               \
        const float* w1 = (const float*)d_in[5 + 6 * L + 0];                   \
        const float* b1 = (const float*)d_in[5 + 6 * L + 1];                   \
        const float* w2 = (const float*)d_in[5 + 6 * L + 2];                   \
        const float* b2 = (const float*)d_in[5 + 6 * L + 3];                   \
        const float* dw = (const float*)d_in[5 + 6 * L + 4];                   \
        const float* db = (const float*)d_in[5 + 6 * L + 5];                   \
        const int hw2   = (HH) * (HH);                                         \
        const size_t sh = (size_t)(NN) * ((CC) + 4 + 32 + 4) * sizeof(float);  \
        fused_pw<CC, MM, NN><<<dim3(hw2 / (NN), 8), 256, sh, stream>>>(        \
            x, w1, b1, w2, b2, zws, hw2);                                      \
        const int tiles = ((HH) + 15) / 16;                                    \
        dwconv21<<<dim3(tiles * tiles, CC, 8), 256, 0, stream>>>(              \
            zws, x, dw, db, (float*)d_out + outOff, CC, HH);                   \
        outOff += (size_t)8 * (CC) * hw2;                                      \
    } while (0)

    RUN_LEVEL(0,   64, 128,  160, 256);  // LDS 106.5 KB
    RUN_LEVEL(1,  128,  64,  320, 128);  // LDS  86.0 KB
    RUN_LEVEL(2,  256,  32,  640,  64);  // LDS  75.8 KB
    RUN_LEVEL(3,  512,  16, 1280,  64);  // LDS 141.4 KB
    RUN_LEVEL(4, 1024,   8, 2560,  32);  // LDS 136.2 KB

#undef RUN_LEVEL
}